// PairCosineSim_31963146616898
// MI455X (gfx1250) — compile-verified
//
#include <hip/hip_runtime.h>

typedef __attribute__((ext_vector_type(2))) float v2f;
typedef __attribute__((ext_vector_type(4))) float v4f;
typedef __attribute__((ext_vector_type(8))) float v8f;

#define BATCH 32
#define TDIM  1024
#define SDIM  1024
#define DDIM  256
#define TILE  64
#define KCHUNK 64
#define LSTR  68          // LDS row stride (floats): 4 mod 64 -> conflict-free frag reads

__global__ __launch_bounds__(256)
void PairCosineSim_kernel(const float* __restrict__ Sup,   // [B, S, D]
                          const float* __restrict__ Xh,    // [B, T, D]
                          float* __restrict__ Out)         // [B, T, S]
{
    __shared__ float Xs[TILE * LSTR];
    __shared__ float Ss[TILE * LSTR];
    __shared__ float xsq[TILE];
    __shared__ float ssq[TILE];
    __shared__ float xinv[TILE];
    __shared__ float sinv[TILE];

    const int b     = blockIdx.z;
    const int tBase = blockIdx.y * TILE;      // rows of X (T dim)
    const int sBase = blockIdx.x * TILE;      // rows of support (S dim)
    const int tid   = threadIdx.x;            // 0..255
    const int lane  = tid & 31;
    const int wave  = tid >> 5;               // 0..7

    if (tid < TILE) { xsq[tid] = 0.0f; ssq[tid] = 0.0f; }

    // Global tile bases
    const float* Xg = Xh  + ((size_t)b * TDIM + tBase) * DDIM;
    const float* Sg = Sup + ((size_t)b * SDIM + sBase) * DDIM;

    // Loader mapping: thread covers rows (tid>>4)+16*i, 16 B-per-thread contiguous float4
    const int lrow  = tid >> 4;               // 0..15
    const int lcol4 = (tid & 15) * 4;         // float4 column within chunk

    // Per-thread running sum-of-squares for the 4 rows this thread loads (free norms)
    float xacc[4] = {0.f, 0.f, 0.f, 0.f};
    float sacc[4] = {0.f, 0.f, 0.f, 0.f};

    // Wave -> 16x32 sub-tile of the 64x64 output tile
    const int m0 = (wave >> 1) * 16;          // row group 0..3
    const int n0 = (wave & 1) * 32;           // col group 0..1 (two 16-wide WMMA tiles)

    v8f acc0 = {};                            // cols [n0,    n0+16)
    v8f acc1 = {};                            // cols [n0+16, n0+32)

    // Fragment addressing (ISA 7.12.2, 32-bit layouts, wave32):
    //   A: lane l holds M = l&15, K = k + 2*(l>>4) .. +1   (2 VGPRs)
    //   B: lane l holds N = l&15, K = k + 2*(l>>4) .. +1   (2 VGPRs)
    const int fr  = lane & 15;                // M for A-frag / N for B-frag
    const int fkb = (lane >> 4) * 2;          // K sub-offset

    for (int kc = 0; kc < DDIM; kc += KCHUNK) {
        __syncthreads();   // previous chunk's fragment reads complete before overwrite

        #pragma unroll
        for (int i = 0; i < 4; ++i) {
            const int r = lrow + 16 * i;
            v4f vx = *(const v4f*)(Xg + (size_t)r * DDIM + kc + lcol4);
            v4f vs = *(const v4f*)(Sg + (size_t)r * DDIM + kc + lcol4);
            *(v4f*)(&Xs[r * LSTR + lcol4]) = vx;
            *(v4f*)(&Ss[r * LSTR + lcol4]) = vs;
            xacc[i] += vx.x * vx.x + vx.y * vx.y + vx.z * vx.z + vx.w * vx.w;
            sacc[i] += vs.x * vs.x + vs.y * vs.y + vs.z * vs.z + vs.w * vs.w;
        }
        __syncthreads();

        #pragma unroll
        for (int ks = 0; ks < KCHUNK; ks += 4) {
            const int kb = ks + fkb;
            v2f a  = *(const v2f*)(&Xs[(m0 + fr)      * LSTR + kb]);
            v2f b0 = *(const v2f*)(&Ss[(n0 + fr)      * LSTR + kb]);
            v2f b1 = *(const v2f*)(&Ss[(n0 + 16 + fr) * LSTR + kb]);
            acc0 = __builtin_amdgcn_wmma_f32_16x16x4_f32(
                false, a, false, b0, (short)0, acc0, false, false);
            acc1 = __builtin_amdgcn_wmma_f32_16x16x4_f32(
                false, a, false, b1, (short)0, acc1, false, false);
        }
    }

    // Reduce row sums-of-squares (ds_add_f32), then build 1/max(norm, eps)
    #pragma unroll
    for (int i = 0; i < 4; ++i) {
        const int r = lrow + 16 * i;
        atomicAdd(&xsq[r], xacc[i]);
        atomicAdd(&ssq[r], sacc[i]);
    }
    __syncthreads();
    if (tid < TILE) {
        xinv[tid] = 1.0f / fmaxf(sqrtf(xsq[tid]), 1e-10f);
        sinv[tid] = 1.0f / fmaxf(sqrtf(ssq[tid]), 1e-10f);
    }
    __syncthreads();

    // Epilogue: C/D layout -> lane l, VGPR v holds M = 8*(l>>4)+v, N = l&15
    const int nlane = lane & 15;
    const int mhalf = (lane >> 4) * 8;
    float* outB = Out + ((size_t)b * TDIM + tBase) * SDIM + sBase;

    #pragma unroll
    for (int v = 0; v < 8; ++v) {
        const int M  = m0 + mhalf + v;
        const float sx = xinv[M];
        const int N0 = n0 + nlane;
        const int N1 = n0 + 16 + nlane;
        outB[(size_t)M * SDIM + N0] = acc0[v] * sx * sinv[N0];
        outB[(size_t)M * SDIM + N1] = acc1[v] * sx * sinv[N1];
    }
}

extern "C" void kernel_launch(void* const* d_in, const int* in_sizes, int n_in,
                              void* d_out, int out_size, void* d_ws, size_t ws_size,
                              hipStream_t stream) {
    (void)in_sizes; (void)n_in; (void)out_size; (void)d_ws; (void)ws_size;
    const float* sup = (const float*)d_in[0];   // support_set [B, S, D]
    const float* xh  = (const float*)d_in[1];   // X_hats      [B, T, D]
    float* out = (float*)d_out;                 // sims        [B, T, S]

    dim3 grid(SDIM / TILE, TDIM / TILE, BATCH); // (16, 16, 32)
    dim3 block(256);
    PairCosineSim_kernel<<<grid, block, 0, stream>>>(sup, xh, out);
}